// R_GAMLP_RLU_17867063951762
// MI455X (gfx1250) — compile-verified
//
#include <hip/hip_runtime.h>

typedef __attribute__((ext_vector_type(2))) float v2f;
typedef __attribute__((ext_vector_type(8))) float v8f;

#define HLAYERS 10
#define NROWS   200000
#define FDIM    128
#define HID     256
#define CDIM    100
#define NEG_SLOPE 0.2f

#define M_TILE    16
#define AS_STRIDE 132   // LDS row stride (132 mod 64 = 4 -> conflict-free fragment reads)
#define HS_STRIDE 516   // LDS row stride for 512-wide concat hidden (516 mod 64 = 4)

__device__ __forceinline__ float leaky_(float x) {
  return (x >= 0.f) ? x : NEG_SLOPE * x;
}

__device__ __forceinline__ float wave_sum32(float x) {
  #pragma unroll
  for (int off = 16; off >= 1; off >>= 1) x += __shfl_xor(x, off, 32);
  return x;
}

__global__ __launch_bounds__(256)
void gamlp_rlu_fused(const float* __restrict__ features,   // (H, N, F)
                     const float* __restrict__ label_emb,  // (N, C)
                     const float* __restrict__ W_att,      // (2F,)
                     const float* __restrict__ b_att,      // scalar
                     const float* __restrict__ Wo1,        // (F, HID)
                     const float* __restrict__ bo1,        // (HID,)
                     const float* __restrict__ ao,         // scalar
                     const float* __restrict__ Wo2,        // (HID, C)
                     const float* __restrict__ bo2,        // (C,)
                     const float* __restrict__ Wl1,        // (C, HID)
                     const float* __restrict__ bl1,        // (HID,)
                     const float* __restrict__ al,         // scalar
                     const float* __restrict__ Wl2,        // (HID, C)
                     const float* __restrict__ bl2,        // (C,)
                     float* __restrict__ out)              // (N, C)
{
  __shared__ float A_s [M_TILE * AS_STRIDE];   // right  (16 x 128)
  __shared__ float L_s [M_TILE * AS_STRIDE];   // label  (16 x 100)
  __shared__ float Hc_s[M_TILE * HS_STRIDE];   // concat hidden (16 x 512)

  const int tid = threadIdx.x;
  // Wave id as a true SGPR so all tile selection / loop control is scalar
  // (keeps EXEC untouched around WMMA and avoids lane-divergent loops).
  const int ws  = __builtin_amdgcn_readfirstlane(tid >> 5);   // 0..7
  const int l   = tid & 31;     // lane id
  const int lc  = l & 15;       // column-in-tile / A row
  const int lh  = l >> 4;       // wave half
  const int rowBase = blockIdx.x * M_TILE;

  // ================= Phase 1: recursive attention -> right (LDS) =================
  const float4 W1v = *(const float4*)(W_att + 4 * l);          // W1[4l..4l+3]
  const float4 W2v = *(const float4*)(W_att + FDIM + 4 * l);   // W2[4l..4l+3]
  const float  batt = b_att[0];

  #pragma unroll
  for (int j = 0; j < 2; ++j) {            // 8 waves x 2 rows = 16 rows
    const int m = ws * 2 + j;
    const int n = rowBase + m;

    float4 f[HLAYERS];
    float  a[HLAYERS], b[HLAYERS];
    #pragma unroll
    for (int i = 0; i < HLAYERS; ++i) {
      f[i] = *(const float4*)(features + ((size_t)i * NROWS + n) * FDIM + 4 * l);
      float pa = f[i].x * W1v.x + f[i].y * W1v.y + f[i].z * W1v.z + f[i].w * W1v.w;
      float pb = f[i].x * W2v.x + f[i].y * W2v.y + f[i].z * W2v.z + f[i].w * W2v.w;
      a[i] = wave_sum32(pa);   // f_i . W1  (broadcast to all lanes)
      b[i] = wave_sum32(pb);   // f_i . W2
    }

    // scalar recursion (redundant per lane; H=10)
    float s[HLAYERS];
    s[0] = leaky_(a[0] + b[0] + batt);
    #pragma unroll
    for (int i = 1; i < HLAYERS; ++i) {
      float mx = s[0];
      #pragma unroll
      for (int k = 1; k < i; ++k) mx = fmaxf(mx, s[k]);
      float se = 0.f, hd = 0.f;
      #pragma unroll
      for (int k = 0; k < i; ++k) {
        float e = __expf(s[k] - mx);
        se += e;
        hd += e * a[k];
      }
      s[i] = leaky_(hd / se + b[i] + batt);
    }

    float mx = s[0];
    #pragma unroll
    for (int i = 1; i < HLAYERS; ++i) mx = fmaxf(mx, s[i]);
    float e[HLAYERS], se = 0.f;
    #pragma unroll
    for (int i = 0; i < HLAYERS; ++i) { e[i] = __expf(s[i] - mx); se += e[i]; }
    const float inv = 1.f / se;

    float4 r; r.x = 0.f; r.y = 0.f; r.z = 0.f; r.w = 0.f;
    #pragma unroll
    for (int i = 0; i < HLAYERS; ++i) {
      const float wi = e[i] * inv;
      r.x += wi * f[i].x; r.y += wi * f[i].y; r.z += wi * f[i].z; r.w += wi * f[i].w;
    }
    *(float4*)(A_s + m * AS_STRIDE + 4 * l) = r;

    if (l < 25)  // 25 lanes * 4 = 100 label cols
      *(float4*)(L_s + m * AS_STRIDE + 4 * l) =
          *(const float4*)(label_emb + (size_t)n * CDIM + 4 * l);
  }
  __syncthreads();

  // ===== Phase 2: hidden = [prelu(right@Wo1+bo1) | prelu(label@Wl1+bl1)] =====
  // Each wave computes two 16x16 tiles per GEMM (cols ws*16 and ws*16+128),
  // sharing one A fragment per k-step between both tiles.
  const int koff = lh * 2;            // K sub-offset per ISA 16x16x4 fragment layout
  const float aoV = ao[0];
  const float alV = al[0];

  { // ---- GEMM1: right (16x128) @ Wo1 (128x256), K compile-time = 128
    const int col0 = ws * 16 + lc;        // hid col of tile ws
    const int col1 = col0 + 128;          // hid col of tile ws+8
    v8f c0 = {}, c1 = {};
    #pragma unroll 8
    for (int k0 = 0; k0 < FDIM; k0 += 4) {
      const int k = k0 + koff;
      v2f av = *(const v2f*)(A_s + lc * AS_STRIDE + k);
      const float* Wr = Wo1 + (size_t)k * HID;
      v2f b0, b1;
      b0.x = Wr[col0];        b1.x = Wr[col1];
      b0.y = Wr[HID + col0];  b1.y = Wr[HID + col1];
      c0 = __builtin_amdgcn_wmma_f32_16x16x4_f32(false, av, false, b0, (short)0, c0, false, false);
      c1 = __builtin_amdgcn_wmma_f32_16x16x4_f32(false, av, false, b1, (short)0, c1, false, false);
    }
    const float bias0 = bo1[col0], bias1 = bo1[col1];
    const int   m0 = lh * 8;
    #pragma unroll
    for (int r = 0; r < 8; ++r) {
      const float x0 = c0[r] + bias0;
      const float x1 = c1[r] + bias1;
      Hc_s[(m0 + r) * HS_STRIDE + col0] = (x0 >= 0.f) ? x0 : aoV * x0;
      Hc_s[(m0 + r) * HS_STRIDE + col1] = (x1 >= 0.f) ? x1 : aoV * x1;
    }
  }

  { // ---- GEMM1b: label (16x100) @ Wl1 (100x256), K compile-time = 100 (25 k-steps)
    const int col0 = ws * 16 + lc;        // col into Wl1 / bl1 (0..255)
    const int col1 = col0 + 128;
    v8f c0 = {}, c1 = {};
    #pragma unroll 5
    for (int k0 = 0; k0 < CDIM; k0 += 4) {
      const int k = k0 + koff;
      v2f av = *(const v2f*)(L_s + lc * AS_STRIDE + k);
      const float* Wr = Wl1 + (size_t)k * HID;
      v2f b0, b1;
      b0.x = Wr[col0];        b1.x = Wr[col1];
      b0.y = Wr[HID + col0];  b1.y = Wr[HID + col1];
      c0 = __builtin_amdgcn_wmma_f32_16x16x4_f32(false, av, false, b0, (short)0, c0, false, false);
      c1 = __builtin_amdgcn_wmma_f32_16x16x4_f32(false, av, false, b1, (short)0, c1, false, false);
    }
    const float bias0 = bl1[col0], bias1 = bl1[col1];
    const int   m0 = lh * 8;
    #pragma unroll
    for (int r = 0; r < 8; ++r) {
      const float x0 = c0[r] + bias0;
      const float x1 = c1[r] + bias1;
      Hc_s[(m0 + r) * HS_STRIDE + HID + col0] = (x0 >= 0.f) ? x0 : alV * x0;
      Hc_s[(m0 + r) * HS_STRIDE + HID + col1] = (x1 >= 0.f) ? x1 : alV * x1;
    }
  }
  __syncthreads();

  // ===== Phase 3: out = hidden(16x512) @ [Wo2;Wl2](512x100) + bo2 + bl2 =====
  if (ws < 7) {                         // scalar branch; waves 0..6 cover 112 cols
    const int col  = ws * 16 + lc;      // 0..111
    const int colc = (col < CDIM) ? col : (CDIM - 1);   // clamp (dropped cols)

    v8f c = {};
    #pragma unroll 8
    for (int k0 = 0; k0 < HID; k0 += 4) {        // Wo2 half, K = 256
      const int k = k0 + koff;
      v2f av = *(const v2f*)(Hc_s + lc * HS_STRIDE + k);
      const float* Wr = Wo2 + (size_t)k * CDIM;
      v2f bv; bv.x = Wr[colc]; bv.y = Wr[CDIM + colc];
      c = __builtin_amdgcn_wmma_f32_16x16x4_f32(false, av, false, bv, (short)0, c, false, false);
    }
    #pragma unroll 8
    for (int k0 = 0; k0 < HID; k0 += 4) {        // Wl2 half, K = 256
      const int k = k0 + koff;
      v2f av = *(const v2f*)(Hc_s + lc * HS_STRIDE + HID + k);
      const float* Wr = Wl2 + (size_t)k * CDIM;
      v2f bv; bv.x = Wr[colc]; bv.y = Wr[CDIM + colc];
      c = __builtin_amdgcn_wmma_f32_16x16x4_f32(false, av, false, bv, (short)0, c, false, false);
    }

    if (col < CDIM) {
      const float bias = bo2[col] + bl2[col];
      const int   m0   = lh * 8;
      #pragma unroll
      for (int r = 0; r < 8; ++r)
        out[(size_t)(rowBase + m0 + r) * CDIM + col] = c[r] + bias;
    }
  }
}

extern "C" void kernel_launch(void* const* d_in, const int* in_sizes, int n_in,
                              void* d_out, int out_size, void* d_ws, size_t ws_size,
                              hipStream_t stream) {
  (void)in_sizes; (void)n_in; (void)out_size; (void)d_ws; (void)ws_size;
  const float* features  = (const float*)d_in[0];
  const float* label_emb = (const float*)d_in[1];
  const float* W_att     = (const float*)d_in[2];
  const float* b_att     = (const float*)d_in[3];
  const float* Wo1       = (const float*)d_in[4];
  const float* bo1       = (const float*)d_in[5];
  const float* ao        = (const float*)d_in[6];
  const float* Wo2       = (const float*)d_in[7];
  const float* bo2       = (const float*)d_in[8];
  const float* Wl1       = (const float*)d_in[9];
  const float* bl1       = (const float*)d_in[10];
  const float* al        = (const float*)d_in[11];
  const float* Wl2       = (const float*)d_in[12];
  const float* bl2       = (const float*)d_in[13];
  float* out = (float*)d_out;

  const int nblocks = NROWS / M_TILE;   // 200000 / 16 = 12500
  gamlp_rlu_fused<<<nblocks, 256, 0, stream>>>(
      features, label_emb, W_att, b_att,
      Wo1, bo1, ao, Wo2, bo2, Wl1, bl1, al, Wl2, bl2, out);
}